// RegionProposalNetwork_61151744360590
// MI455X (gfx1250) — compile-verified
//
#include <hip/hip_runtime.h>
#include <cstdint>
#include <cstddef>

// ---------------- problem constants (from reference) ----------------
#define NM   8                    // N_MESH
#define AA   4                    // anchors per cell
#define NB6  6                    // box params
#define N0   1048576              // 64^3 * 4
#define N1   131072               // 32^3 * 4
#define N2   16384                // 16^3 * 4
#define WHD0 262144               // 2^18
#define WHD1 32768                // 2^15
#define WHD2 4096                 // 2^12
#define BINS 8192
#define BIN_SHIFT 19              // key >> 19 -> 13-bit bin
#define CAP  8192                 // candidate capacity per (batch,level)
#define CLIPV 4.135166556742356f  // log(1000/16)
#define MESHSZ 160.0f

typedef int v4i __attribute__((ext_vector_type(4)));

// monotone float -> u32 key (order-preserving, incl. negatives)
__device__ __forceinline__ uint32_t fkey(float f) {
  uint32_t b = __float_as_uint(f);
  uint32_t m = (uint32_t)(-(int32_t)(b >> 31)) | 0x80000000u;
  return b ^ m;
}
__device__ __forceinline__ float fkey_inv(uint32_t k) {
  uint32_t b = (k & 0x80000000u) ? (k ^ 0x80000000u) : ~k;
  return __uint_as_float(b);
}

// ---------------- CDNA5 async global->LDS path ----------------
template <int N>
__device__ __forceinline__ void wait_async() {
#if __has_builtin(__builtin_amdgcn_s_wait_asynccnt)
  __builtin_amdgcn_s_wait_asynccnt(N);
#else
  asm volatile("s_wait_asynccnt %0" :: "i"(N) : "memory");
#endif
}

__device__ __forceinline__ void async_g2l_b128(const void* g, void* l) {
#if __has_builtin(__builtin_amdgcn_global_load_async_to_lds_b128)
  __builtin_amdgcn_global_load_async_to_lds_b128(
      (__attribute__((address_space(1))) v4i*)(size_t)g,
      (__attribute__((address_space(3))) v4i*)(uint32_t)(size_t)l,
      0, 0);
#else
  uint32_t lo = (uint32_t)(size_t)l;            // low 32 bits == LDS offset
  unsigned long long ga = (unsigned long long)(size_t)g;
  asm volatile("global_load_async_to_lds_b128 %0, %1, off"
               :: "v"(lo), "v"(ga) : "memory");
#endif
}

// ---------------- K1: streaming histogram (async double-buffered) ----------------
__global__ __launch_bounds__(256)
void hist_kernel(const float* __restrict__ cls, uint32_t* __restrict__ hist,
                 int n_per_batch, int blocks_per_batch, int level)
{
  __shared__ uint32_t lh[BINS];       // 32 KB
  __shared__ float4 stage[2][256];    // 8 KB staging, per-thread slots
  for (int i = threadIdx.x; i < BINS; i += 256) lh[i] = 0;
  __syncthreads();

  const int b     = blockIdx.x / blocks_per_batch;
  const int chunk = blockIdx.x % blocks_per_batch;
  const int elems = n_per_batch / blocks_per_batch;      // 16384
  const float* src = cls + (size_t)b * n_per_batch + (size_t)chunk * elems;
  const int ntiles = elems >> 10;                        // 1024 floats / tile

  async_g2l_b128(src + threadIdx.x * 4, &stage[0][threadIdx.x]);
  for (int t = 0; t < ntiles; ++t) {
    const int cur = t & 1;
    if (t + 1 < ntiles) {
      async_g2l_b128(src + (size_t)(t + 1) * 1024 + threadIdx.x * 4,
                     &stage[cur ^ 1][threadIdx.x]);
      wait_async<1>();      // oldest (tile t) complete; newest in flight
    } else {
      wait_async<0>();
    }
    float4 v = stage[cur][threadIdx.x];
    atomicAdd(&lh[fkey(v.x) >> BIN_SHIFT], 1u);
    atomicAdd(&lh[fkey(v.y) >> BIN_SHIFT], 1u);
    atomicAdd(&lh[fkey(v.z) >> BIN_SHIFT], 1u);
    atomicAdd(&lh[fkey(v.w) >> BIN_SHIFT], 1u);
  }
  __syncthreads();

  uint32_t* gh = hist + (size_t)(b * 3 + level) * BINS;
  for (int i = threadIdx.x; i < BINS; i += 256) {
    uint32_t c = lh[i];
    if (c) atomicAdd(&gh[i], c);
  }
}

// ---------------- K2: suffix scan of histogram -> threshold bin ----------------
__global__ __launch_bounds__(256)
void scan_kernel(const uint32_t* __restrict__ hist, uint32_t* __restrict__ thr,
                 const int* __restrict__ pre)
{
  const int id = blockIdx.x;          // b*3 + l
  const int l  = id % 3;
  const int nl = (l == 0) ? N0 : (l == 1) ? N1 : N2;
  const uint32_t k = (uint32_t)min(pre[0], nl);

  __shared__ uint32_t s[BINS];        // reversed (highest bin first)
  __shared__ uint32_t tsum[256];
  __shared__ uint32_t found;
  const uint32_t* h = hist + (size_t)id * BINS;
  for (int i = threadIdx.x; i < BINS; i += 256) s[i] = h[BINS - 1 - i];
  if (threadIdx.x == 0) found = BINS;
  __syncthreads();

  const int base = threadIdx.x * 32;  // 256 * 32 == 8192
  uint32_t run = 0;
  for (int q = 0; q < 32; ++q) { run += s[base + q]; s[base + q] = run; }
  tsum[threadIdx.x] = run;
  __syncthreads();
  if (threadIdx.x == 0) {
    uint32_t acc = 0;
    for (int q = 0; q < 256; ++q) { uint32_t t = tsum[q]; tsum[q] = acc; acc += t; }
  }
  __syncthreads();
  const uint32_t off = tsum[threadIdx.x];
  for (int q = 0; q < 32; ++q) {
    if (off + s[base + q] >= k) { atomicMin(&found, (uint32_t)(base + q)); break; }
  }
  __syncthreads();
  if (threadIdx.x == 0) {
    uint32_t i = (found < BINS) ? found : (BINS - 1);
    thr[id] = (uint32_t)(BINS - 1 - i);   // candidate iff bin >= thr
  }
}

// ---------------- K3: compaction of candidates (bin >= threshold) ----------------
__global__ __launch_bounds__(256)
void compact_kernel(const float* __restrict__ cls, const uint32_t* __restrict__ thr,
                    uint32_t* __restrict__ cnt, unsigned long long* __restrict__ cand,
                    int n_per_batch, int blocks_per_batch, int level, int whd_log2)
{
  const int b     = blockIdx.x / blocks_per_batch;
  const int chunk = blockIdx.x % blocks_per_batch;
  const int elems = n_per_batch / blocks_per_batch;
  const size_t base = (size_t)b * n_per_batch + (size_t)chunk * elems;
  const uint32_t tb = thr[b * 3 + level];
  const float4* src = (const float4*)(cls + base);
  const int n4 = elems >> 2;
  uint32_t* c = cnt + (b * 3 + level);
  unsigned long long* cb = cand + (size_t)(b * 3 + level) * CAP;
  const uint32_t whd_mask = (1u << whd_log2) - 1u;

  for (int i = threadIdx.x; i < n4; i += 256) {
    const float4 v = src[i];
    const int j0 = chunk * elems + i * 4;   // index within this batch
    const float vv[4] = {v.x, v.y, v.z, v.w};
#pragma unroll
    for (int q = 0; q < 4; ++q) {
      const uint32_t key = fkey(vv[q]);
      if ((key >> BIN_SHIFT) >= tb) {
        const uint32_t j = (uint32_t)(j0 + q);
        const uint32_t a = j >> whd_log2;      // channel
        const uint32_t sp = j & whd_mask;      // spatial
        const uint32_t t = (sp << 2) | a;      // permuted flat index
        const uint32_t pos = atomicAdd(c, 1u);
        if (pos < CAP)
          cb[pos] = ((unsigned long long)key << 32) | (uint32_t)(t ^ 0xFFFFFFFFu);
      }
    }
  }
}

// ---------------- K4: bitonic sort (desc) + sigmoid + decode + clip ----------------
__global__ __launch_bounds__(1024)
void sortout_kernel(const unsigned long long* __restrict__ cand,
                    const uint32_t* __restrict__ cnt, const int* __restrict__ pre,
                    const float* __restrict__ reg0, const float* __restrict__ reg1,
                    const float* __restrict__ reg2, const float* __restrict__ anchors,
                    float* __restrict__ out)
{
  __shared__ unsigned long long arr[CAP];  // 64 KB
  const int id = blockIdx.x;               // b*3 + l
  const int b = id / 3;
  const int l = id % 3;

  const int p  = pre[0];
  const int k0 = min(p, N0), k1 = min(p, N1), k2v = min(p, N2);
  const int ktot = k0 + k1 + k2v;
  const int k    = (l == 0) ? k0 : (l == 1) ? k1 : k2v;
  const int col0 = (l == 0) ? 0 : (l == 1) ? k0 : (k0 + k1);

  uint32_t c = cnt[id]; if (c > CAP) c = CAP;
  const unsigned long long* cb = cand + (size_t)id * CAP;
  for (int i = threadIdx.x; i < CAP; i += blockDim.x)
    arr[i] = (i < (int)c) ? cb[i] : 0ull;

  // bitonic sort, descending (key-major, lower original index first on ties)
  for (unsigned sz = 2; sz <= CAP; sz <<= 1) {
    for (unsigned st = sz >> 1; st > 0; st >>= 1) {
      __syncthreads();
      for (unsigned i = threadIdx.x; i < CAP; i += blockDim.x) {
        const unsigned pr = i ^ st;
        if (pr > i) {
          const unsigned long long a = arr[i], bb = arr[pr];
          const bool descBlock = ((i & sz) == 0);
          if (descBlock ? (a < bb) : (a > bb)) { arr[i] = bb; arr[pr] = a; }
        }
      }
    }
  }
  __syncthreads();

  const float* reg  = (l == 0) ? reg0 : (l == 1) ? reg1 : reg2;
  const int    WHD  = (l == 0) ? WHD0 : (l == 1) ? WHD1 : WHD2;
  const size_t aoff = (l == 0) ? 0 : (l == 1) ? (size_t)N0 : (size_t)(N0 + N1);
  float* scores = out;
  float* props  = out + (size_t)NM * ktot;

  for (int j = threadIdx.x; j < k; j += blockDim.x) {
    const unsigned long long e = arr[j];
    const uint32_t key = (uint32_t)(e >> 32);
    uint32_t t = ((uint32_t)e) ^ 0xFFFFFFFFu;
    if (e == 0ull) t = 0;                       // padding guard
    const float v = fkey_inv(key);
    const float sc = 1.0f / (1.0f + expf(-v));
    const int col = col0 + j;
    scores[(size_t)b * ktot + col] = sc;

    // gather deltas: source channel = a*6 + c, spatial s, batch stride 24*WHD
    const uint32_t sp = t >> 2, a = t & 3u;
    const float* rp = reg + ((size_t)b * (AA * NB6) + (size_t)a * NB6) * (size_t)WHD + sp;
    const float d0 = rp[0];
    const float d1 = rp[(size_t)1 * WHD];
    const float d2 = rp[(size_t)2 * WHD];
    const float d3 = rp[(size_t)3 * WHD];
    const float d4 = rp[(size_t)4 * WHD];
    const float d5 = rp[(size_t)5 * WHD];

    const float* ap = anchors + ((size_t)aoff + t) * 6;
    const float s0 = ap[3] - ap[0], s1 = ap[4] - ap[1], s2 = ap[5] - ap[2];
    const float c0 = ap[0] + 0.5f * s0, c1 = ap[1] + 0.5f * s1, c2 = ap[2] + 0.5f * s2;
    const float pc0 = d0 * s0 + c0, pc1 = d1 * s1 + c1, pc2 = d2 * s2 + c2;
    const float e0 = expf(fminf(d3, CLIPV)) * s0;
    const float e1 = expf(fminf(d4, CLIPV)) * s1;
    const float e2 = expf(fminf(d5, CLIPV)) * s2;

    float* pb = props + ((size_t)b * ktot + col) * 6;
    pb[0] = fminf(fmaxf(pc0 - 0.5f * e0, 0.0f), MESHSZ);
    pb[1] = fminf(fmaxf(pc1 - 0.5f * e1, 0.0f), MESHSZ);
    pb[2] = fminf(fmaxf(pc2 - 0.5f * e2, 0.0f), MESHSZ);
    pb[3] = fminf(fmaxf(pc0 + 0.5f * e0, 0.0f), MESHSZ);
    pb[4] = fminf(fmaxf(pc1 + 0.5f * e1, 0.0f), MESHSZ);
    pb[5] = fminf(fmaxf(pc2 + 0.5f * e2, 0.0f), MESHSZ);
  }
}

// ---------------- host launcher ----------------
extern "C" void kernel_launch(void* const* d_in, const int* in_sizes, int n_in,
                              void* d_out, int out_size, void* d_ws, size_t ws_size,
                              hipStream_t stream) {
  (void)in_sizes; (void)n_in; (void)out_size; (void)ws_size;
  const float* cls0 = (const float*)d_in[0];
  const float* reg0 = (const float*)d_in[1];
  const float* cls1 = (const float*)d_in[2];
  const float* reg1 = (const float*)d_in[3];
  const float* cls2 = (const float*)d_in[4];
  const float* reg2 = (const float*)d_in[5];
  const float* anchors = (const float*)d_in[6];
  const int*   pre  = (const int*)d_in[7];

  uint8_t* ws = (uint8_t*)d_ws;
  uint32_t* hist = (uint32_t*)ws;                       // 24 * BINS u32
  const size_t off_thr  = (size_t)24 * BINS * sizeof(uint32_t);
  uint32_t* thr = (uint32_t*)(ws + off_thr);            // 24 u32
  const size_t off_cnt  = off_thr + 128;
  uint32_t* cnt = (uint32_t*)(ws + off_cnt);            // 24 u32
  const size_t off_cand = off_cnt + 128;
  unsigned long long* cand = (unsigned long long*)(ws + off_cand); // 24*CAP u64

  (void)hipMemsetAsync(d_ws, 0, off_cand, stream);

  hist_kernel<<<dim3(NM * 64), dim3(256), 0, stream>>>(cls0, hist, N0, 64, 0);
  hist_kernel<<<dim3(NM * 8),  dim3(256), 0, stream>>>(cls1, hist, N1, 8, 1);
  hist_kernel<<<dim3(NM * 1),  dim3(256), 0, stream>>>(cls2, hist, N2, 1, 2);

  scan_kernel<<<dim3(24), dim3(256), 0, stream>>>(hist, thr, pre);

  compact_kernel<<<dim3(NM * 64), dim3(256), 0, stream>>>(cls0, thr, cnt, cand, N0, 64, 0, 18);
  compact_kernel<<<dim3(NM * 8),  dim3(256), 0, stream>>>(cls1, thr, cnt, cand, N1, 8, 1, 15);
  compact_kernel<<<dim3(NM * 1),  dim3(256), 0, stream>>>(cls2, thr, cnt, cand, N2, 1, 2, 12);

  sortout_kernel<<<dim3(24), dim3(1024), 0, stream>>>(cand, cnt, pre,
                                                      reg0, reg1, reg2, anchors,
                                                      (float*)d_out);
}